// char_matchingV2_61203283968670
// MI455X (gfx1250) — compile-verified
//
#include <hip/hip_runtime.h>
#include <hip/hip_bf16.h>
#include <stdint.h>

#define B_ 8
#define P_ 512
#define G_ 128
#define C_ 68
#define MAXV 8

#if defined(__has_builtin)
#  if __has_builtin(__builtin_amdgcn_tensor_load_to_lds)
#    define HAVE_TDM 1
#  endif
#endif
#ifndef HAVE_TDM
#  define HAVE_TDM 0
#endif

typedef __attribute__((ext_vector_type(4))) unsigned int u32x4;
typedef __attribute__((ext_vector_type(8))) int          i32x8;
typedef __attribute__((ext_vector_type(4))) int          i32x4;

__device__ __forceinline__ float cross2(float ux, float uy, float vx, float vy) {
#pragma clang fp contract(off)
  return ux * vy - uy * vx;
}

__device__ __forceinline__ bool quad_degenerate(const float x[4], const float y[4]) {
  bool d = false;
#pragma unroll
  for (int i = 0; i < 4; ++i)
#pragma unroll
    for (int j = i + 1; j < 4; ++j)
      d = d || ((x[i] == x[j]) && (y[i] == y[j]));
  return d;
}

// Sutherland-Hodgman: area of intersection of convex quads (mirrors reference).
__device__ float quad_isect_area(const float sx[4], const float sy[4],
                                 const float cxi[4], const float cyi[4]) {
#pragma clang fp contract(off)
  // orient clip CCW (reference flips when signed area < 0)
  float sa = 0.f;
#pragma unroll
  for (int i = 0; i < 4; ++i) {
    int j = (i + 1) & 3;
    sa += cross2(cxi[i], cyi[i], cxi[j], cyi[j]);
  }
  const bool rev = (sa < 0.f);
  float cx[4], cy[4];
#pragma unroll
  for (int i = 0; i < 4; ++i) {
    int k = rev ? (3 - i) : i;
    cx[i] = cxi[k];
    cy[i] = cyi[k];
  }

  float px[MAXV], py[MAXV];
#pragma unroll
  for (int i = 0; i < 4; ++i) { px[i] = sx[i]; py[i] = sy[i]; }
#pragma unroll
  for (int i = 4; i < MAXV; ++i) { px[i] = 0.f; py[i] = 0.f; }
  int n = 4;

  for (int e = 0; e < 4; ++e) {
    const float ax = cx[e], ay = cy[e];
    const float bx = cx[(e + 1) & 3], by = cy[(e + 1) & 3];
    const float ex = bx - ax, ey = by - ay;
    float ox[MAXV], oy[MAXV];
    int m = 0;
    for (int idx = 0; idx < MAXV; ++idx) {
      if (idx >= n) break;
      const int nx = (idx + 1 >= n) ? 0 : idx + 1;
      const float dp = cross2(ex, ey, px[idx] - ax, py[idx] - ay);
      const float dq = cross2(ex, ey, px[nx] - ax, py[nx] - ay);
      const bool inp = (dp >= 0.f);
      const bool inq = (dq >= 0.f);
      if (inp && (m < MAXV)) { ox[m] = px[idx]; oy[m] = py[idx]; ++m; }
      if (inp != inq) {
        const float denom = dp - dq;
        const float t = (denom == 0.f) ? 0.f : (dp / denom);
        const float ipx = px[idx] + t * (px[nx] - px[idx]);
        const float ipy = py[idx] + t * (py[nx] - py[idx]);
        if (m < MAXV) { ox[m] = ipx; oy[m] = ipy; ++m; }
      }
    }
    n = m;
    for (int i = 0; i < MAXV; ++i) {
      if (i < n) { px[i] = ox[i]; py[i] = oy[i]; }
    }
  }

  float area = 0.f;
  for (int idx = 0; idx < MAXV; ++idx) {
    if (idx >= n) break;
    const int nx = (idx + 1 >= n) ? 0 : idx + 1;
    area += cross2(px[idx], py[idx], px[nx], py[nx]);
  }
  return fabsf(0.5f * area);
}

// Kernel 1: one block per (b,g); one thread per pred. TDM stages the batch's
// 512 pred quads (8-of-9-float strided rows) into LDS, then LDS argmax.
__global__ __launch_bounds__(P_) void area_argmax_kernel(
    const float* __restrict__ bboxes,   // [B,P,9]
    const float* __restrict__ polys,    // [B,G,4,2]
    int* __restrict__ match_idx) {      // [B*G]
  __shared__ float s_pred[P_ * 8];
  __shared__ float s_val[P_];
  __shared__ int   s_ix[P_];

  const int bg = blockIdx.x;
  const int b  = bg >> 7;  // G_ == 128
  const int t  = threadIdx.x;

  const float* gbase = bboxes + (size_t)b * (P_ * 9);

#if HAVE_TDM
  if (t < 32) {  // wave 0 issues the DMA (TDM ignores EXEC; wave-level op)
    const unsigned long long ga = (unsigned long long)(uintptr_t)gbase;
    const unsigned lds_off = (unsigned)(uintptr_t)(void*)s_pred;  // low 32 bits = LDS offset
    // D# group0: count=1 | lds_addr | global_addr[56:0] | type=2
    u32x4 g0 = { 1u,
                 lds_off,
                 (unsigned)(ga & 0xFFFFFFFFull),
                 (unsigned)((ga >> 32) & 0x1FFFFFFull) | (2u << 30) };
    // D# group1: data_size=4B; tensor 9 x 512 (stride 9); tile 8 x 512
    i32x8 g1 = { (int)(2u << 16),                       // data_size=2 (4B)
                 (int)(9u << 16),                       // tensor_dim0 lo -> bits 63:48
                 (int)(((unsigned)P_) << 16),           // tensor_dim1 lo -> bits 95:80
                 (int)(8u << 16),                       // tile_dim0 -> bits 127:112
                 (int)P_,                               // tile_dim1 -> bits 143:128
                 9,                                     // tensor_dim0_stride lo
                 (int)(((9u * (unsigned)P_) & 0xFFFFu) << 16),  // dim1_stride lo
                 (int)((9u * (unsigned)P_) >> 16) };
    i32x4 z4 = { 0, 0, 0, 0 };
#if __has_include(<hip/amd_detail/amd_gfx1250_TDM.h>)
    i32x8 z8 = { 0, 0, 0, 0, 0, 0, 0, 0 };
    __builtin_amdgcn_tensor_load_to_lds(g0, g1, z4, z4, z8, 0);
#else
    __builtin_amdgcn_tensor_load_to_lds(g0, g1, z4, z4, 0);
#endif
#if __has_builtin(__builtin_amdgcn_s_wait_tensorcnt)
    __builtin_amdgcn_s_wait_tensorcnt(0);
#else
    asm volatile("s_wait_tensorcnt 0x0" ::: "memory");
#endif
  }
#else
  for (int i = t; i < P_ * 8; i += P_) {
    const int p = i >> 3, k = i & 7;
    s_pred[i] = gbase[p * 9 + k];
  }
#endif
  __syncthreads();

  // clip (GT) quad: uniform across block
  const float* cp = polys + (size_t)bg * 8;
  float cx[4], cy[4];
#pragma unroll
  for (int i = 0; i < 4; ++i) { cx[i] = cp[2 * i]; cy[i] = cp[2 * i + 1]; }

  // this thread's pred quad from LDS
  float sxv[4], syv[4];
#pragma unroll
  for (int i = 0; i < 4; ++i) {
    sxv[i] = s_pred[t * 8 + 2 * i];
    syv[i] = s_pred[t * 8 + 2 * i + 1];
  }

  float area = quad_isect_area(sxv, syv, cx, cy);
  if (quad_degenerate(sxv, syv)) area = 0.f;  // reference zeroes degenerate preds

  s_val[t] = area;
  s_ix[t]  = t;
  __syncthreads();
  for (int s = P_ / 2; s > 0; s >>= 1) {
    if (t < s) {
      const float v2 = s_val[t + s];
      const int   i2 = s_ix[t + s];
      if (v2 > s_val[t] || (v2 == s_val[t] && i2 < s_ix[t])) {
        s_val[t] = v2;
        s_ix[t]  = i2;
      }
    }
    __syncthreads();
  }
  if (t == 0) {
    const bool gdeg    = quad_degenerate(cx, cy);
    const bool matched = (!gdeg) && (s_val[0] != 0.f);
    match_idx[bg] = matched ? s_ix[0] : -1;
  }
}

// Kernel 2: per (b,g) cross-entropy + correctness (deterministic, no atomics).
__global__ __launch_bounds__(128) void ce_kernel(
    const float* __restrict__ scores,    // [B,P,C]
    const int* __restrict__ labels,      // [B*G]
    const int* __restrict__ match_idx,   // [B*G]
    float* __restrict__ ce_arr,          // [B*G]
    int* __restrict__ corr_arr) {        // [B*G]
  __shared__ float s_v[128];
  __shared__ int   s_i[128];
  __shared__ float s_sum[128];
  __shared__ float s_lab;

  const int bg    = blockIdx.x;
  const int b     = bg >> 7;
  const int t     = threadIdx.x;
  const int m     = match_idx[bg];
  const int label = labels[bg];

  const float* row = scores + ((size_t)b * P_ + (size_t)(m < 0 ? 0 : m)) * C_;
  if (t == 0 && m >= 0) __builtin_prefetch(row, 0, 0);  // global_prefetch_b8

  const bool active = (t < C_);
  float sc = 0.f;
  if (active && m >= 0) sc = row[t];   // unmatched -> zero scores (matches ref)
  if (t == label) s_lab = sc;

  s_v[t] = active ? sc : -3.402823466e38f;
  s_i[t] = active ? t : (C_ + t);
  __syncthreads();
  for (int s = 64; s > 0; s >>= 1) {
    if (t < s) {
      const float v2 = s_v[t + s];
      const int   i2 = s_i[t + s];
      if (v2 > s_v[t] || (v2 == s_v[t] && i2 < s_i[t])) {
        s_v[t] = v2;
        s_i[t] = i2;
      }
    }
    __syncthreads();
  }
  const float maxv = s_v[0];
  const int   amax = s_i[0];

  s_sum[t] = active ? expf(sc - maxv) : 0.f;
  __syncthreads();
  for (int s = 64; s > 0; s >>= 1) {
    if (t < s) s_sum[t] += s_sum[t + s];
    __syncthreads();
  }

  if (t == 0) {
    ce_arr[bg]   = maxv + logf(s_sum[0]) - s_lab;   // CE = logsumexp - s[label]
    corr_arr[bg] = (m >= 0 && amax == label) ? 1 : 0;
  }
}

// Kernel 3: fixed-order final reduction -> d_out
__global__ __launch_bounds__(512) void finalize_kernel(
    const float* __restrict__ ce_arr,
    const int* __restrict__ corr_arr,
    float* __restrict__ out) {
  __shared__ float sv[512];
  __shared__ int   si[512];
  const int t = threadIdx.x;
  sv[t] = ce_arr[t] + ce_arr[t + 512];
  si[t] = corr_arr[t] + corr_arr[t + 512];
  __syncthreads();
  for (int s = 256; s > 0; s >>= 1) {
    if (t < s) {
      sv[t] += sv[t + s];
      si[t] += si[t + s];
    }
    __syncthreads();
  }
  if (t == 0) {
    out[0] = sv[0] / (float)(B_ * G_);  // loss
    out[1] = (float)(B_ * G_);          // total_number
    out[2] = (float)si[0];              // rec_correct
  }
}

extern "C" void kernel_launch(void* const* d_in, const int* in_sizes, int n_in,
                              void* d_out, int out_size, void* d_ws, size_t ws_size,
                              hipStream_t stream) {
  (void)in_sizes; (void)n_in; (void)out_size; (void)ws_size;
  const float* bboxes = (const float*)d_in[0];  // [B,P,9]
  const float* scores = (const float*)d_in[1];  // [B,P,C]
  const float* polys  = (const float*)d_in[2];  // [B,G,4,2]
  const int*   labels = (const int*)d_in[3];    // [B,G]

  int*   match_idx = (int*)d_ws;
  float* ce_arr    = (float*)((char*)d_ws + 4096);
  int*   corr_arr  = (int*)((char*)d_ws + 8192);

  area_argmax_kernel<<<B_ * G_, P_, 0, stream>>>(bboxes, polys, match_idx);
  ce_kernel<<<B_ * G_, 128, 0, stream>>>(scores, labels, match_idx, ce_arr, corr_arr);
  finalize_kernel<<<1, 512, 0, stream>>>(ce_arr, corr_arr, (float*)d_out);
}